// GAT_50680614092808
// MI455X (gfx1250) — compile-verified
//
#include <hip/hip_runtime.h>
#include <hip/hip_bf16.h>
#include <cmath>

// ---------------------------------------------------------------------------
// GAT (2-layer) for MI455X / gfx1250, wave32.
// Dense GEMM h1 = x @ W1 done with V_WMMA_F32_16X16X4_F32 (fp32-exact).
// W1 staged in LDS in K-pair-interleaved layout -> each B fragment is one
// ds_load_b64. Main K-loop fully unguarded; single branch-free tail step.
// Edge softmax/aggregation via L2 atomics (max / add), vectorized loads.
// ---------------------------------------------------------------------------

typedef __attribute__((ext_vector_type(2))) float v2f;
typedef __attribute__((ext_vector_type(8))) float v8f;

#define F_IN 165
#define KPAIRS 84     // ceil(168/2) K-pairs (K padded 165 -> 168)
#define HD1  64       // 8 heads x 8 dims
#define NHEAD1 8
#define NEG_SLOPE 0.2f

// ---- float atomic max via integer bit tricks ------------------------------
__device__ __forceinline__ void atomic_max_f32(float* addr, float val) {
    if (val >= 0.0f) {
        atomicMax((int*)addr, __float_as_int(val));
    } else {
        atomicMin((unsigned int*)addr, __float_as_uint(val));
    }
}

__device__ __forceinline__ void edge_sd(const long long* __restrict__ ei,
                                        long long E, long long e,
                                        long long& s, long long& d) {
    if (e < E) { s = ei[e]; d = ei[E + e]; }
    else       { s = e - E; d = s; }          // self loop
}

// ---- generic fill ---------------------------------------------------------
__global__ void fill_kernel(float* __restrict__ p, float v, long long n) {
    long long i = (long long)blockIdx.x * blockDim.x + threadIdx.x;
    if (i < n) p[i] = v;
}

// ---- K1: h1 = x @ W1 via WMMA f32 16x16x4 ---------------------------------
// One wave -> 16 nodes x 64 features (4 accumulators). Block = 4 waves.
// LDS layout: ldsW[kp*128 + n*2 + j] = W1[2*kp + j][n]  (zero-padded K tail)
__global__ void gemm1_wmma(const float* __restrict__ x,
                           const float* __restrict__ W1,
                           float* __restrict__ h1, int n_tiles) {
    __shared__ float ldsW[KPAIRS * 128];  // 43008 bytes
    for (int i = threadIdx.x; i < KPAIRS * 64; i += blockDim.x) {
        int kp = i >> 6, n = i & 63;
        int k0 = 2 * kp;
        ldsW[kp * 128 + n * 2 + 0] = (k0     < F_IN) ? W1[k0 * HD1 + n]       : 0.0f;
        ldsW[kp * 128 + n * 2 + 1] = (k0 + 1 < F_IN) ? W1[(k0 + 1) * HD1 + n] : 0.0f;
    }
    __syncthreads();

    int wave = threadIdx.x >> 5;
    int tile = blockIdx.x * 4 + wave;
    if (tile >= n_tiles) return;          // wave-uniform: EXEC stays all-1s

    int lane = threadIdx.x & 31;
    int half = lane >> 4;                 // 0: K pair {k,k+1}, 1: {k+2,k+3}
    int lq   = lane & 15;                 // M (for A) / N (for B,C)
    const float* xrow = x + (size_t)(tile * 16 + lq) * F_IN;

    v8f c0 = {}, c1 = {}, c2 = {}, c3 = {};

    // Main loop: K = 0..163, 41 fully in-bounds steps (ka <= 162, ka+1 <= 163)
    for (int k = 0; k < 164; k += 4) {
        int ka = k + 2 * half;
        v2f a;
        a.x = xrow[ka];
        a.y = xrow[ka + 1];

        int kp = (ka >> 1);               // = k/2 + half
        const float* bp = &ldsW[kp * 128 + lq * 2];
        v2f b0 = *(const v2f*)(bp + 0);
        v2f b1 = *(const v2f*)(bp + 32);
        v2f b2 = *(const v2f*)(bp + 64);
        v2f b3 = *(const v2f*)(bp + 96);

        c0 = __builtin_amdgcn_wmma_f32_16x16x4_f32(false, a, false, b0, (short)0, c0, false, false);
        c1 = __builtin_amdgcn_wmma_f32_16x16x4_f32(false, a, false, b1, (short)0, c1, false, false);
        c2 = __builtin_amdgcn_wmma_f32_16x16x4_f32(false, a, false, b2, (short)0, c2, false, false);
        c3 = __builtin_amdgcn_wmma_f32_16x16x4_f32(false, a, false, b3, (short)0, c3, false, false);
    }

    // Branch-free tail: covers K = 164..167 (only K=164 is real).
    // xrow[164] is in-bounds for every lane; cndmask zeroes the half==1 lanes.
    {
        float v = xrow[F_IN - 1];                   // K = 164
        v2f a;
        a.x = (half == 0) ? v : 0.0f;
        a.y = 0.0f;
        int kp = 82 + half;                         // pairs (164,165) / (166,167)
        const float* bp = &ldsW[kp * 128 + lq * 2]; // LDS already zero-padded
        v2f b0 = *(const v2f*)(bp + 0);
        v2f b1 = *(const v2f*)(bp + 32);
        v2f b2 = *(const v2f*)(bp + 64);
        v2f b3 = *(const v2f*)(bp + 96);

        c0 = __builtin_amdgcn_wmma_f32_16x16x4_f32(false, a, false, b0, (short)0, c0, false, false);
        c1 = __builtin_amdgcn_wmma_f32_16x16x4_f32(false, a, false, b1, (short)0, c1, false, false);
        c2 = __builtin_amdgcn_wmma_f32_16x16x4_f32(false, a, false, b2, (short)0, c2, false, false);
        c3 = __builtin_amdgcn_wmma_f32_16x16x4_f32(false, a, false, b3, (short)0, c3, false, false);
    }

    // C layout: VGPR r -> row M = r + 8*half, col = lq
    int node0 = tile * 16 + 8 * half;
    #pragma unroll
    for (int r = 0; r < 8; ++r) {
        float* hr = h1 + (size_t)(node0 + r) * HD1 + lq;
        hr[0]  = c0[r];
        hr[16] = c1[r];
        hr[32] = c2[r];
        hr[48] = c3[r];
    }
}

// ---- K2: per-(node,head) attention projections for layer 1 ----------------
__global__ void alphas1_kernel(const float* __restrict__ h1,
                               const float* __restrict__ a_src1,
                               const float* __restrict__ a_dst1,
                               float* __restrict__ as1, float* __restrict__ ad1,
                               long long n_nodes) {
    long long i = (long long)blockIdx.x * blockDim.x + threadIdx.x;
    if (i >= n_nodes * NHEAD1) return;
    long long n = i >> 3;
    int h = (int)(i & 7);
    const float* hp = h1 + n * HD1 + h * 8;
    const float* sp = a_src1 + h * 8;
    const float* dp = a_dst1 + h * 8;
    float as = 0.f, ad = 0.f;
    #pragma unroll
    for (int j = 0; j < 8; ++j) { float v = hp[j]; as += v * sp[j]; ad += v * dp[j]; }
    as1[i] = as; ad1[i] = ad;
}

// ---- K3: layer-1 segment max over dst -------------------------------------
__global__ void edge_max1(const long long* __restrict__ ei, long long E, long long Etot,
                          const float* __restrict__ as1, const float* __restrict__ ad1,
                          float* __restrict__ m1) {
    long long e = (long long)blockIdx.x * blockDim.x + threadIdx.x;
    if (e >= Etot) return;
    long long s, d; edge_sd(ei, E, e, s, d);
    const float* asp = as1 + s * 8;
    const float* adp = ad1 + d * 8;
    #pragma unroll
    for (int h = 0; h < 8; ++h) {
        float el = asp[h] + adp[h];
        el = (el > 0.f) ? el : NEG_SLOPE * el;
        atomic_max_f32(&m1[d * 8 + h], el);
    }
}

// ---- K4: layer-1 softmax denominator --------------------------------------
__global__ void edge_sum1(const long long* __restrict__ ei, long long E, long long Etot,
                          const float* __restrict__ as1, const float* __restrict__ ad1,
                          const float* __restrict__ m1, float* __restrict__ z1) {
    long long e = (long long)blockIdx.x * blockDim.x + threadIdx.x;
    if (e >= Etot) return;
    long long s, d; edge_sd(ei, E, e, s, d);
    const float* asp = as1 + s * 8;
    const float* adp = ad1 + d * 8;
    const float* mp  = m1  + d * 8;
    #pragma unroll
    for (int h = 0; h < 8; ++h) {
        float el = asp[h] + adp[h];
        el = (el > 0.f) ? el : NEG_SLOPE * el;
        atomicAdd(&z1[d * 8 + h], __expf(el - mp[h]));
    }
}

// ---- K5: layer-1 weighted aggregation (one thread per edge,head) ----------
__global__ void edge_agg1(const long long* __restrict__ ei, long long E, long long Etot,
                          const float* __restrict__ as1, const float* __restrict__ ad1,
                          const float* __restrict__ m1, const float* __restrict__ z1,
                          const float* __restrict__ h1, float* __restrict__ out1) {
    long long i = (long long)blockIdx.x * blockDim.x + threadIdx.x;
    if (i >= Etot * 8) return;
    long long e = i >> 3;
    int h = (int)(i & 7);
    long long s, d; edge_sd(ei, E, e, s, d);
    float el = as1[s * 8 + h] + ad1[d * 8 + h];
    el = (el > 0.f) ? el : NEG_SLOPE * el;
    float alpha = __expf(el - m1[d * 8 + h]) / z1[d * 8 + h];
    const float* hp = h1 + s * HD1 + h * 8;
    float* op = out1 + d * HD1 + h * 8;
    #pragma unroll
    for (int j = 0; j < 8; ++j) atomicAdd(&op[j], alpha * hp[j]);
}

// ---- K6: bias + ELU + h2 = out1 @ W2 + layer-2 attention projections ------
__global__ void node_mid(const float* __restrict__ out1acc, const float* __restrict__ b1,
                         const float* __restrict__ W2,
                         const float* __restrict__ a_src2, const float* __restrict__ a_dst2,
                         float* __restrict__ h2, float* __restrict__ as2,
                         float* __restrict__ ad2, long long n_nodes) {
    long long n = (long long)blockIdx.x * blockDim.x + threadIdx.x;
    if (n >= n_nodes) return;
    const float* op = out1acc + n * HD1;
    float acc0 = 0.f, acc1 = 0.f;
    #pragma unroll 8
    for (int j = 0; j < HD1; ++j) {
        float v = op[j] + b1[j];
        v = (v > 0.f) ? v : expm1f(v);    // ELU
        acc0 += v * W2[j * 2 + 0];
        acc1 += v * W2[j * 2 + 1];
    }
    h2[n * 2 + 0] = acc0;
    h2[n * 2 + 1] = acc1;
    as2[n] = acc0 * a_src2[0] + acc1 * a_src2[1];
    ad2[n] = acc0 * a_dst2[0] + acc1 * a_dst2[1];
}

// ---- K7/K8/K9: layer-2 (single head, dim 2) edge passes -------------------
__global__ void edge_max2(const long long* __restrict__ ei, long long E, long long Etot,
                          const float* __restrict__ as2, const float* __restrict__ ad2,
                          float* __restrict__ m2) {
    long long e = (long long)blockIdx.x * blockDim.x + threadIdx.x;
    if (e >= Etot) return;
    long long s, d; edge_sd(ei, E, e, s, d);
    float el = as2[s] + ad2[d];
    el = (el > 0.f) ? el : NEG_SLOPE * el;
    atomic_max_f32(&m2[d], el);
}

__global__ void edge_sum2(const long long* __restrict__ ei, long long E, long long Etot,
                          const float* __restrict__ as2, const float* __restrict__ ad2,
                          const float* __restrict__ m2, float* __restrict__ z2) {
    long long e = (long long)blockIdx.x * blockDim.x + threadIdx.x;
    if (e >= Etot) return;
    long long s, d; edge_sd(ei, E, e, s, d);
    float el = as2[s] + ad2[d];
    el = (el > 0.f) ? el : NEG_SLOPE * el;
    atomicAdd(&z2[d], __expf(el - m2[d]));
}

__global__ void edge_agg2(const long long* __restrict__ ei, long long E, long long Etot,
                          const float* __restrict__ as2, const float* __restrict__ ad2,
                          const float* __restrict__ m2, const float* __restrict__ z2,
                          const float* __restrict__ h2, float* __restrict__ out2) {
    long long e = (long long)blockIdx.x * blockDim.x + threadIdx.x;
    if (e >= Etot) return;
    long long s, d; edge_sd(ei, E, e, s, d);
    float el = as2[s] + ad2[d];
    el = (el > 0.f) ? el : NEG_SLOPE * el;
    float alpha = __expf(el - m2[d]) / z2[d];
    atomicAdd(&out2[d * 2 + 0], alpha * h2[s * 2 + 0]);
    atomicAdd(&out2[d * 2 + 1], alpha * h2[s * 2 + 1]);
}

// ---- K10: bias + log_softmax over 2 classes -------------------------------
__global__ void final_kernel(const float* __restrict__ out2, const float* __restrict__ b2,
                             float* __restrict__ out, long long n_nodes) {
    long long n = (long long)blockIdx.x * blockDim.x + threadIdx.x;
    if (n >= n_nodes) return;
    float o0 = out2[n * 2 + 0] + b2[0];   // mean over 1 head == identity
    float o1 = out2[n * 2 + 1] + b2[1];
    float mm = fmaxf(o0, o1);
    float lse = mm + logf(__expf(o0 - mm) + __expf(o1 - mm));
    out[n * 2 + 0] = o0 - lse;
    out[n * 2 + 1] = o1 - lse;
}

// ---------------------------------------------------------------------------
extern "C" void kernel_launch(void* const* d_in, const int* in_sizes, int n_in,
                              void* d_out, int out_size, void* d_ws, size_t ws_size,
                              hipStream_t stream) {
    const float*     x      = (const float*)d_in[0];
    const long long* ei     = (const long long*)d_in[1];   // int64 edge_index [2,E]
    const float*     W1     = (const float*)d_in[2];
    const float*     a_src1 = (const float*)d_in[3];
    const float*     a_dst1 = (const float*)d_in[4];
    const float*     b1     = (const float*)d_in[5];
    const float*     W2     = (const float*)d_in[6];
    const float*     a_src2 = (const float*)d_in[7];
    const float*     a_dst2 = (const float*)d_in[8];
    const float*     b2     = (const float*)d_in[9];
    float*           out    = (float*)d_out;

    const long long N    = in_sizes[0] / F_IN;   // 100000
    const long long E    = in_sizes[1] / 2;      // 1600000
    const long long Etot = E + N;                // + self loops

    // workspace layout (floats); [m1,m2] then [z1,out1,z2,out2] contiguous
    float* w    = (float*)d_ws;
    float* h1   = w;  w += N * 64;
    float* as1  = w;  w += N * 8;
    float* ad1  = w;  w += N * 8;
    float* h2   = w;  w += N * 2;
    float* as2  = w;  w += N;
    float* ad2  = w;  w += N;
    float* m1   = w;  w += N * 8;   // -inf region start
    float* m2   = w;  w += N;
    float* z1   = w;  w += N * 8;   // zero region start
    float* out1 = w;  w += N * 64;
    float* z2   = w;  w += N;
    float* out2 = w;  w += N * 2;

    const int B = 256;
    long long ninf_cnt = N * 9;
    long long zero_cnt = N * (8 + 64 + 1 + 2);
    fill_kernel<<<(int)((ninf_cnt + B - 1) / B), B, 0, stream>>>(m1, -INFINITY, ninf_cnt);
    fill_kernel<<<(int)((zero_cnt + B - 1) / B), B, 0, stream>>>(z1, 0.0f, zero_cnt);

    int n_tiles = (int)((N + 15) / 16);          // 6250 tiles of 16 nodes
    gemm1_wmma<<<(n_tiles + 3) / 4, 128, 0, stream>>>(x, W1, h1, n_tiles);

    alphas1_kernel<<<(int)((N * 8 + B - 1) / B), B, 0, stream>>>(h1, a_src1, a_dst1, as1, ad1, N);

    int gE  = (int)((Etot + B - 1) / B);
    int gE8 = (int)((Etot * 8 + B - 1) / B);
    edge_max1<<<gE,  B, 0, stream>>>(ei, E, Etot, as1, ad1, m1);
    edge_sum1<<<gE,  B, 0, stream>>>(ei, E, Etot, as1, ad1, m1, z1);
    edge_agg1<<<gE8, B, 0, stream>>>(ei, E, Etot, as1, ad1, m1, z1, h1, out1);

    node_mid<<<(int)((N + B - 1) / B), B, 0, stream>>>(out1, b1, W2, a_src2, a_dst2,
                                                       h2, as2, ad2, N);

    edge_max2<<<gE, B, 0, stream>>>(ei, E, Etot, as2, ad2, m2);
    edge_sum2<<<gE, B, 0, stream>>>(ei, E, Etot, as2, ad2, m2, z2);
    edge_agg2<<<gE, B, 0, stream>>>(ei, E, Etot, as2, ad2, m2, z2, h2, out2);

    final_kernel<<<(int)((N + B - 1) / B), B, 0, stream>>>(out2, b2, out, N);
}